// _GEN_NETWORK_70858370450108
// MI455X (gfx1250) — compile-verified
//
#include <hip/hip_runtime.h>
#include <hip/hip_bf16.h>
#include <math.h>

// Sizes from the reference
#define V_ 32000
#define E_ 512
#define H_ 512
#define L_ 512
#define K_ 128
#define B_ 64
#define T_ 64

typedef __attribute__((ext_vector_type(2))) float v2f;
typedef __attribute__((ext_vector_type(8))) float v8f;

// ---------------------------------------------------------------------------
// Narrow fp32 WMMA GEMM (per-step GEMMs):  C[M,N] = A[M,K] * op(W) + bias
//   TRANS=true : W is [N,K] row-major (A @ W.T);  false: W is [K,N] (A @ W)
// blockDim = 128 (4 waves), wave w owns rows [blockIdx.y*64+16w, +16),
// blockIdx.x owns 16 columns. Weight tile staged in LDS.
// ---------------------------------------------------------------------------
template <bool TRANS>
__global__ void gemm_wmma_f32(const float* __restrict__ A, const float* __restrict__ W,
                              const float* __restrict__ bias, float* __restrict__ C,
                              int M, int N, int K) {
  constexpr int KC = 64;
  __shared__ float sB[16][KC + 1];

  const int lane = threadIdx.x & 31;
  const int wave = threadIdx.x >> 5;
  const int half = lane >> 4;   // 0: K pair {0,1}; 1: K pair {2,3}
  const int nl   = lane & 15;   // A row / B col within tile
  const int colBase = blockIdx.x * 16;
  const int rowBase = blockIdx.y * 64 + wave * 16;

  const float* Arow = A + (size_t)(rowBase + nl) * K;

  v8f acc = {};
  for (int k0 = 0; k0 < K; k0 += KC) {
    __syncthreads();
    for (int i = threadIdx.x; i < 16 * KC; i += blockDim.x) {
      int n = i / KC, kk = i % KC;
      sB[n][kk] = TRANS ? W[(size_t)(colBase + n) * K + (k0 + kk)]
                        : W[(size_t)(k0 + kk) * N + (colBase + n)];
    }
    __syncthreads();
#pragma unroll
    for (int kk = 0; kk < KC; kk += 4) {
      v2f a, b;
      a.x = Arow[k0 + kk + 2 * half];
      a.y = Arow[k0 + kk + 2 * half + 1];
      b.x = sB[nl][kk + 2 * half];
      b.y = sB[nl][kk + 2 * half + 1];
      acc = __builtin_amdgcn_wmma_f32_16x16x4_f32(false, a, false, b,
                                                  (short)0, acc, false, false);
    }
  }

  float bb = bias ? bias[colBase + nl] : 0.0f;
#pragma unroll
  for (int v = 0; v < 8; v++) {
    int row = rowBase + v + half * 8;
    C[(size_t)row * N + colBase + nl] = acc[v] + bb;
  }
}

// ---------------------------------------------------------------------------
// Wide fp32 WMMA GEMM for the dominant logits projection (A @ W.T + bias).
// Each wave computes 16 rows x 64 cols (4 accumulators) so every A fragment
// feeds 4 WMMAs (4x less A traffic). The 64x64 weight tile (16 KB) is staged
// into LDS with CDNA5 async copies (GLOBAL_LOAD_ASYNC_TO_LDS_B128, ASYNCcnt).
// LDS rows padded to 68 floats: 272 B row stride keeps 16 B alignment for the
// B128 async stores and makes fragment reads bank-conflict free.
// Requires K % 64 == 0, N % 64 == 0, M % 64 == 0.
// ---------------------------------------------------------------------------
__global__ void gemm_wmma_f32_wide(const float* __restrict__ A, const float* __restrict__ W,
                                   const float* __restrict__ bias, float* __restrict__ C,
                                   int M, int N, int K) {
  constexpr int KC = 64;
  constexpr int NC = 64;
  constexpr int LDB = KC + 4;          // 68 floats, 272 B row stride
  __shared__ float sB[NC][LDB];

  const int lane = threadIdx.x & 31;
  const int wave = threadIdx.x >> 5;
  const int half = lane >> 4;
  const int nl   = lane & 15;
  const int colBase = blockIdx.x * NC;
  const int rowBase = blockIdx.y * 64 + wave * 16;

  const float*   Arow    = A + (size_t)(rowBase + nl) * K;
  const uint32_t sB_base = (uint32_t)(uintptr_t)&sB[0][0];  // wave-relative LDS addr

  v8f acc[4] = {{}, {}, {}, {}};

  for (int k0 = 0; k0 < K; k0 += KC) {
    __syncthreads();
    // Stage 64 cols x 64 K floats = 1024 float4 with async LDS copies:
    // each of 128 threads issues 8 x B128 (16 B) transfers.
#pragma unroll
    for (int it = 0; it < 8; it++) {
      const int idx = it * 128 + threadIdx.x;
      const int n = idx >> 4;              // 0..63 (column within tile)
      const int q = (idx & 15) << 2;       // 0,4,...,60 (K offset, float4)
      const float* gsrc = W + (size_t)(colBase + n) * K + (k0 + q);
      const uint32_t ldst = sB_base + (uint32_t)((n * LDB + q) * 4);
      asm volatile("global_load_async_to_lds_b128 %0, %1, off"
                   :: "v"(ldst), "v"(gsrc) : "memory");
    }
    asm volatile("s_wait_asynccnt 0x0" ::: "memory");
    __syncthreads();

#pragma unroll
    for (int kk = 0; kk < KC; kk += 4) {
      v2f a;
      a.x = Arow[k0 + kk + 2 * half];
      a.y = Arow[k0 + kk + 2 * half + 1];
#pragma unroll
      for (int c = 0; c < 4; c++) {
        v2f b;
        b.x = sB[c * 16 + nl][kk + 2 * half];
        b.y = sB[c * 16 + nl][kk + 2 * half + 1];
        acc[c] = __builtin_amdgcn_wmma_f32_16x16x4_f32(false, a, false, b,
                                                       (short)0, acc[c], false, false);
      }
    }
  }

#pragma unroll
  for (int c = 0; c < 4; c++) {
    const int col = colBase + c * 16 + nl;
    const float bb = bias ? bias[col] : 0.0f;
#pragma unroll
    for (int v = 0; v < 8; v++) {
      int row = rowBase + v + half * 8;
      C[(size_t)row * N + col] = acc[c][v] + bb;
    }
  }
}

// ---------------------------------------------------------------------------
// Setup: gather prob / i_hid, u_hid = prob @ user_emb.T, uih = u_hid + i_hid,
// zero h0. One block per batch row, 128 threads.
// ---------------------------------------------------------------------------
__global__ void setup_kernel(const int* __restrict__ user_ids, const int* __restrict__ item_ids,
                             const float* __restrict__ ucp, const float* __restrict__ item_emb,
                             const float* __restrict__ user_emb,
                             float* __restrict__ prob, float* __restrict__ i_hid,
                             float* __restrict__ uih, float* __restrict__ h0) {
  const int b = blockIdx.x;
  const int tid = threadIdx.x;  // 128
  __shared__ float sp[K_];

  float p = ucp[(size_t)user_ids[b] * K_ + tid];
  prob[b * K_ + tid] = p;
  sp[tid] = p;
  __syncthreads();

  const float* irow = item_emb + (size_t)item_ids[b] * L_;
  for (int l = tid; l < L_; l += 128) {
    float ih = irow[l];
    i_hid[b * L_ + l] = ih;
    const float* ue = user_emb + (size_t)l * K_;
    float uh = 0.f;
#pragma unroll 8
    for (int k = 0; k < K_; k++) uh += sp[k] * ue[k];
    uih[b * L_ + l] = uh + ih;
    h0[b * H_ + l] = 0.f;
  }
}

// ---------------------------------------------------------------------------
// inp = emb[seq[:, t]] + ui     (64 blocks x 128 threads)
// ---------------------------------------------------------------------------
__global__ void make_inp(const float* __restrict__ emb, const int* __restrict__ seq,
                         const float* __restrict__ ui, float* __restrict__ inp, int t) {
  const int b = blockIdx.x;
  const int idx = seq[b * T_ + t];
  const float* er = emb + (size_t)idx * E_;
  const float* ur = ui + (size_t)b * E_;
  float* o = inp + (size_t)b * E_;
  for (int e = threadIdx.x; e < E_; e += blockDim.x) o[e] = er[e] + ur[e];
}

// ---------------------------------------------------------------------------
// Fused GRU step (WMMA): gi = inp@W_ih.T, gh = h@W_hh.T for one 16-column
// tile (3 gates each -> 6 accumulators), gate fusion, writes h_out and
// outs[:, t, :].  Grid: 32 blocks (H/16) x 128 threads.
// ---------------------------------------------------------------------------
__global__ void gru_step_wmma(const float* __restrict__ inp, const float* __restrict__ hin,
                              const float* __restrict__ Wih, const float* __restrict__ Whh,
                              const float* __restrict__ bih, const float* __restrict__ bhh,
                              float* __restrict__ hout, float* __restrict__ outs, int t) {
  constexpr int KC = 64;
  __shared__ float sW[6][16][KC + 1];  // g<3: W_ih gate g; g>=3: W_hh gate g-3

  const int lane = threadIdx.x & 31;
  const int wave = threadIdx.x >> 5;
  const int half = lane >> 4;
  const int nl   = lane & 15;
  const int colBase = blockIdx.x * 16;  // column tile in H
  const int rowBase = wave * 16;        // 4 waves cover B=64 rows

  const float* Ai = inp + (size_t)(rowBase + nl) * E_;
  const float* Ah = hin + (size_t)(rowBase + nl) * H_;

  v8f gi[3] = {{}, {}, {}};
  v8f gh[3] = {{}, {}, {}};

  for (int k0 = 0; k0 < 512; k0 += KC) {
    __syncthreads();
    for (int i = threadIdx.x; i < 6 * 16 * KC; i += blockDim.x) {
      int g = i / (16 * KC);
      int r = i % (16 * KC);
      int n = r / KC, kk = r % KC;
      int col = (g % 3) * 512 + colBase + n;  // gate row in [3H, 512] weight
      const float* Wsrc = (g < 3) ? Wih : Whh;
      sW[g][n][kk] = Wsrc[(size_t)col * 512 + (k0 + kk)];
    }
    __syncthreads();
#pragma unroll
    for (int kk = 0; kk < KC; kk += 4) {
      v2f ai, ah;
      ai.x = Ai[k0 + kk + 2 * half];
      ai.y = Ai[k0 + kk + 2 * half + 1];
      ah.x = Ah[k0 + kk + 2 * half];
      ah.y = Ah[k0 + kk + 2 * half + 1];
#pragma unroll
      for (int g = 0; g < 3; g++) {
        v2f bi, bh;
        bi.x = sW[g][nl][kk + 2 * half];
        bi.y = sW[g][nl][kk + 2 * half + 1];
        bh.x = sW[g + 3][nl][kk + 2 * half];
        bh.y = sW[g + 3][nl][kk + 2 * half + 1];
        gi[g] = __builtin_amdgcn_wmma_f32_16x16x4_f32(false, ai, false, bi,
                                                      (short)0, gi[g], false, false);
        gh[g] = __builtin_amdgcn_wmma_f32_16x16x4_f32(false, ah, false, bh,
                                                      (short)0, gh[g], false, false);
      }
    }
  }

  const int col = colBase + nl;
  const float br_i = bih[col], bz_i = bih[512 + col], bn_i = bih[1024 + col];
  const float br_h = bhh[col], bz_h = bhh[512 + col], bn_h = bhh[1024 + col];
#pragma unroll
  for (int v = 0; v < 8; v++) {
    int row = rowBase + v + half * 8;  // batch index
    float hold = hin[(size_t)row * H_ + col];
    float r = 1.f / (1.f + expf(-(gi[0][v] + br_i + gh[0][v] + br_h)));
    float z = 1.f / (1.f + expf(-(gi[1][v] + bz_i + gh[1][v] + bz_h)));
    float n = tanhf(gi[2][v] + bn_i + r * (gh[2][v] + bn_h));
    float hn = (1.f - z) * n + z * hold;
    hout[(size_t)row * H_ + col] = hn;
    outs[((size_t)row * T_ + t) * H_ + col] = hn;
  }
}

// ---------------------------------------------------------------------------
// Per-row: item_score = dot(a, i_hid); softmax over [us_raw*prob, item_score];
// ui_mid = (s[:128]*prob) @ user_emb.T + s[128]*i_hid.   64 blocks x 128 thr.
// ---------------------------------------------------------------------------
__global__ void softmax_ui(const float* __restrict__ a, const float* __restrict__ i_hid,
                           const float* __restrict__ prob, const float* __restrict__ us_raw,
                           const float* __restrict__ user_emb, float* __restrict__ ui_mid) {
  const int b = blockIdx.x;
  const int tid = threadIdx.x;  // 128
  __shared__ float red[128];
  __shared__ float w[128];

  // item score
  float part = 0.f;
  for (int l = tid; l < L_; l += 128) part += a[b * L_ + l] * i_hid[b * L_ + l];
  red[tid] = part;
  __syncthreads();
  for (int s = 64; s > 0; s >>= 1) {
    if (tid < s) red[tid] += red[tid + s];
    __syncthreads();
  }
  const float iscore = red[0];
  __syncthreads();

  const float us = us_raw[b * K_ + tid] * prob[b * K_ + tid];

  // max over 129
  red[tid] = us;
  __syncthreads();
  for (int s = 64; s > 0; s >>= 1) {
    if (tid < s) red[tid] = fmaxf(red[tid], red[tid + s]);
    __syncthreads();
  }
  const float mx = fmaxf(red[0], iscore);
  __syncthreads();

  // sum of exp
  const float e = expf(us - mx);
  red[tid] = e;
  __syncthreads();
  for (int s = 64; s > 0; s >>= 1) {
    if (tid < s) red[tid] += red[tid + s];
    __syncthreads();
  }
  const float ei = expf(iscore - mx);
  const float denom = red[0] + ei;
  const float si = ei / denom;
  w[tid] = (e / denom) * prob[b * K_ + tid];
  __syncthreads();

  for (int l = tid; l < L_; l += 128) {
    float acc = si * i_hid[b * L_ + l];
    const float* ue = user_emb + (size_t)l * K_;
#pragma unroll 8
    for (int k = 0; k < K_; k++) acc += w[k] * ue[k];
    ui_mid[b * L_ + l] = acc;
  }
}

// ---------------------------------------------------------------------------
extern "C" void kernel_launch(void* const* d_in, const int* in_sizes, int n_in,
                              void* d_out, int out_size, void* d_ws, size_t ws_size,
                              hipStream_t stream) {
  const int*   seq      = (const int*)d_in[0];
  const int*   user_ids = (const int*)d_in[1];
  const int*   item_ids = (const int*)d_in[2];
  // d_in[3]: random_flag (unused)
  const float* emb      = (const float*)d_in[4];
  const float* user_emb = (const float*)d_in[5];
  const float* ucp      = (const float*)d_in[6];
  const float* item_emb = (const float*)d_in[7];
  const float* l2o_W    = (const float*)d_in[8];
  const float* l2o_b    = (const float*)d_in[9];
  const float* attn_W   = (const float*)d_in[10];
  const float* attn_b   = (const float*)d_in[11];
  const float* W_ih     = (const float*)d_in[12];
  const float* W_hh     = (const float*)d_in[13];
  const float* b_ih     = (const float*)d_in[14];
  const float* b_hh     = (const float*)d_in[15];
  const float* out_W    = (const float*)d_in[16];
  const float* out_b    = (const float*)d_in[17];
  float* out = (float*)d_out;

  // Workspace layout (floats)
  float* ws     = (float*)d_ws;
  float* prob   = ws; ws += B_ * K_;       // [64,128]
  float* i_hid  = ws; ws += B_ * L_;       // [64,512]
  float* uih    = ws; ws += B_ * L_;       // u_hid + i_hid
  float* ui     = ws; ws += B_ * E_;       // recurrent ui
  float* h0buf  = ws; ws += B_ * H_;
  float* h1buf  = ws; ws += B_ * H_;
  float* inp    = ws; ws += B_ * E_;
  float* abuf   = ws; ws += B_ * L_;
  float* us_raw = ws; ws += B_ * K_;
  float* ui_mid = ws; ws += B_ * L_;
  float* outs   = ws; ws += B_ * T_ * H_;  // [64,64,512] = 8 MB

  // Setup: gathers + u_hid GEMV + h0 = 0, then ui0 = uih @ l2o_W.T + l2o_b
  setup_kernel<<<B_, 128, 0, stream>>>(user_ids, item_ids, ucp, item_emb, user_emb,
                                       prob, i_hid, uih, h0buf);
  gemm_wmma_f32<true><<<dim3(E_ / 16, 1), 128, 0, stream>>>(uih, l2o_W, l2o_b, ui,
                                                            B_, E_, L_);

  float* hb[2] = {h0buf, h1buf};
  for (int t = 0; t < T_; t++) {
    float* h_in  = hb[t & 1];
    float* h_out = hb[(t + 1) & 1];
    make_inp<<<B_, 128, 0, stream>>>(emb, seq, ui, inp, t);
    gru_step_wmma<<<H_ / 16, 128, 0, stream>>>(inp, h_in, W_ih, W_hh, b_ih, b_hh,
                                               h_out, outs, t);
    // a = h @ attn_W.T + attn_b
    gemm_wmma_f32<true><<<dim3(L_ / 16, 1), 128, 0, stream>>>(h_out, attn_W, attn_b,
                                                              abuf, B_, L_, H_);
    // us_raw = a @ user_emb    (user_emb is [L,K] row-major, direct)
    gemm_wmma_f32<false><<<dim3(K_ / 16, 1), 128, 0, stream>>>(abuf, user_emb, nullptr,
                                                               us_raw, B_, K_, L_);
    softmax_ui<<<B_, 128, 0, stream>>>(abuf, i_hid, prob, us_raw, user_emb, ui_mid);
    // ui = ui_mid @ l2o_W.T + l2o_b
    gemm_wmma_f32<true><<<dim3(E_ / 16, 1), 128, 0, stream>>>(ui_mid, l2o_W, l2o_b,
                                                              ui, B_, E_, L_);
  }

  // logits[4096, 32000] = outs @ out_W.T + out_b   (dominant GEMM, 134 GFLOP)
  // Wide kernel: 4x A reuse + async weight staging into LDS.
  gemm_wmma_f32_wide<<<dim3(V_ / 64, (B_ * T_) / 64), 128, 0, stream>>>(
      outs, out_W, out_b, out, B_ * T_, V_, H_);
}